// AttentionLayer_2130303779016
// MI455X (gfx1250) — compile-verified
//
#include <hip/hip_runtime.h>

typedef float v2f __attribute__((ext_vector_type(2)));
typedef float v4f __attribute__((ext_vector_type(4)));
typedef float v8f __attribute__((ext_vector_type(8)));

#define B_  16
#define TE_ 512
#define TD_ 64
#define HE_ 512
#define HD_ 512
#define LD_ 512   // every matrix here has leading dim 512; K is always 512

// ---- fast tanh: gfx1250 has V_TANH_F32 (transcendental) ----
#if __has_builtin(__builtin_amdgcn_tanhf)
__device__ __forceinline__ float fast_tanh(float x) { return __builtin_amdgcn_tanhf(x); }
#elif __has_builtin(__builtin_amdgcn_tanh_f32)
__device__ __forceinline__ float fast_tanh(float x) { return __builtin_amdgcn_tanh_f32(x); }
#else
__device__ __forceinline__ float fast_tanh(float x) { return tanhf(x); }
#endif

__device__ __forceinline__ float wave_sum(float v) {
#pragma unroll
  for (int off = 16; off > 0; off >>= 1) v += __shfl_xor(v, off, 32);
  return v;
}
__device__ __forceinline__ float wave_max(float v) {
#pragma unroll
  for (int off = 16; off > 0; off >>= 1) v = fmaxf(v, __shfl_xor(v, off, 32));
  return v;
}

// Only enforce builtin presence in the DEVICE compile pass (host pass lacks it).
#if defined(__HIP_DEVICE_COMPILE__) && !__has_builtin(__builtin_amdgcn_wmma_f32_16x16x4_f32)
#error "wmma f32 16x16x4 builtin missing on this toolchain (device pass)"
#endif

// ============================================================================
// Register-blocked fp32 WMMA GEMM, specialized to LD=512 / K=512:
//   C[M,512] = A[M,512] * B[512,512]  (row-major, batched via blockIdx.z)
// One wave computes a 64x32 macro-tile = 4 M-tiles x 2 N-tiles of 16x16.
// Per K-step (4): 4x global_load_b64 (A) + 4x b32 (B) feed 8 WMMAs
// -> 1 vmem per v_wmma_f32_16x16x4_f32, with shift/immediate addressing.
//
// ISA 7.12.2 layouts (wave32):
//   A 16x4:   lane l holds A[m = l&15][k = 2*(l>>4) + v], v = VGPR index 0..1
//   B 4x16:   lane l holds B[k = 2*(l>>4) + v][n = l&15]
//   C/D:      VGPR r: lanes 0-15 -> row r, lanes 16-31 -> row r+8, col = l&15
// ============================================================================
__global__ __launch_bounds__(32) void wmma_gemm512_kernel(
    const float* __restrict__ A, const float* __restrict__ Bm,
    float* __restrict__ C, long long sA, long long sB, long long sC)
{
  const int lane = threadIdx.x & 31;
  const int half = lane >> 4;   // 0 or 1
  const int l16  = lane & 15;
  const int n0 = blockIdx.x * 32;
  const int m0 = blockIdx.y * 64;
  const float* Ab = A  + (long long)blockIdx.z * sA;
  const float* Bb = Bm + (long long)blockIdx.z * sB;
  float*       Cb = C  + (long long)blockIdx.z * sC;

  const float* arow0 = Ab + (long long)(m0 + l16) * LD_ + 2 * half; // K contiguous
  const float* bcol0 = Bb + (long long)(2 * half) * LD_ + n0 + l16; // K strided

#if __has_builtin(__builtin_amdgcn_wmma_f32_16x16x4_f32)
  v8f acc[4][2] = {};

#pragma unroll 4
  for (int k = 0; k < 512; k += 4) {
    v2f a[4];
    v2f b[2];
#pragma unroll
    for (int i = 0; i < 4; ++i)     // 4 M-tiles, b64 each
      a[i] = *(const v2f*)(arow0 + (long long)(i * 16) * LD_ + k);
#pragma unroll
    for (int j = 0; j < 2; ++j) {   // 2 N-tiles, 2x b32 each
      b[j].x = bcol0[(long long)k * LD_ + j * 16];        // B[k+2h  ][n]
      b[j].y = bcol0[(long long)(k + 1) * LD_ + j * 16];  // B[k+2h+1][n]
    }
#pragma unroll
    for (int i = 0; i < 4; ++i)
#pragma unroll
      for (int j = 0; j < 2; ++j)
        // 8 args: (neg_a, A, neg_b, B, c_mod, C, reuse_a, reuse_b)
        acc[i][j] = __builtin_amdgcn_wmma_f32_16x16x4_f32(
            false, a[i], false, b[j], (short)0, acc[i][j], false, false);
  }

#pragma unroll
  for (int i = 0; i < 4; ++i) {
    float* crow = Cb + (long long)(m0 + 16 * i + 8 * half) * LD_ + n0 + l16;
#pragma unroll
    for (int j = 0; j < 2; ++j)
#pragma unroll
      for (int r = 0; r < 8; ++r)
        crow[(long long)r * LD_ + 16 * j] = acc[i][j][r];
  }
#else
  // Host-pass / fallback body (never the device path on this toolchain):
  // same 64x32 macro-tile computed with scalar FMAs, identical data layout.
  (void)arow0; (void)bcol0;
  float accs[4][2][8] = {};
  for (int k = 0; k < 512; ++k) {
#pragma unroll
    for (int i = 0; i < 4; ++i)
#pragma unroll
      for (int j = 0; j < 2; ++j) {
        float bv = Bb[(long long)k * LD_ + n0 + 16 * j + l16];
#pragma unroll
        for (int r = 0; r < 8; ++r)
          accs[i][j][r] += Ab[(long long)(m0 + 16 * i + 8 * half + r) * LD_ + k] * bv;
      }
  }
#pragma unroll
  for (int i = 0; i < 4; ++i) {
    float* crow = Cb + (long long)(m0 + 16 * i + 8 * half) * LD_ + n0 + l16;
#pragma unroll
    for (int j = 0; j < 2; ++j)
#pragma unroll
      for (int r = 0; r < 8; ++r)
        crow[(long long)r * LD_ + 16 * j] = accs[i][j][r];
  }
#endif
}

// ============================================================================
// Fused logits + softmax:
//   e[b,d,t] = softmax_t( sum_h V[h] * tanh(Ws[b,t,h] + Uh[b,d,h]) )
// One 512-thread block (16 waves) per (d, b). Wave w handles t = 32w..32w+31.
// Lanes cover h = 4*lane + 128*j (j=0..3): 4x global_load_b128 per Ws row,
// Uh/V slices in per-lane registers, shuffle reduce, block softmax over 512.
// ============================================================================
__global__ __launch_bounds__(512) void logits_softmax_kernel(
    const float* __restrict__ Ws,   // [B, TE, HE]
    const float* __restrict__ Uh,   // [B, TD, HE]
    const float* __restrict__ Va,   // [HE]
    float* __restrict__ e_out)      // [B, TD, TE]
{
  __shared__ float s_logit[TE_];
  __shared__ float s_part[16];
  __shared__ float s_bcast;

  const int d = blockIdx.x;
  const int b = blockIdx.y;
  const int tid  = threadIdx.x;
  const int w    = tid >> 5;
  const int lane = tid & 31;

  // per-lane slices of Uh[b,d,:] and V at h = 4*lane + 128*j (b128 loads)
  v4f uh[4], vv[4];
  const float* uhrow = Uh + ((long long)b * TD_ + d) * HE_;
#pragma unroll
  for (int j = 0; j < 4; ++j) {
    int h = 4 * lane + 128 * j;
    uh[j] = *(const v4f*)(uhrow + h);
    vv[j] = *(const v4f*)(Va + h);
  }

  const float* wsb = Ws + (long long)b * TE_ * HE_;
#pragma unroll 2
  for (int i = 0; i < 32; ++i) {
    int t = w * 32 + i;
    const float* row = wsb + (long long)t * HE_ + 4 * lane;
    float acc = 0.f;
#pragma unroll
    for (int j = 0; j < 4; ++j) {
      v4f wsv = *(const v4f*)(row + 128 * j);
#pragma unroll
      for (int c = 0; c < 4; ++c)
        acc += vv[j][c] * fast_tanh(wsv[c] + uh[j][c]);
    }
    acc = wave_sum(acc);
    if (lane == 0) s_logit[t] = acc;
  }
  __syncthreads();

  // ---- softmax over TE=512 (one element per thread) ----
  float x = s_logit[tid];
  float m = wave_max(x);
  if (lane == 0) s_part[w] = m;
  __syncthreads();
  if (tid == 0) {
    float mm = s_part[0];
#pragma unroll
    for (int i = 1; i < 16; ++i) mm = fmaxf(mm, s_part[i]);
    s_bcast = mm;
  }
  __syncthreads();
  float ex = __expf(x - s_bcast);
  float sm = wave_sum(ex);
  if (lane == 0) s_part[w] = sm;
  __syncthreads();
  if (tid == 0) {
    float ss = 0.f;
#pragma unroll
    for (int i = 0; i < 16; ++i) ss += s_part[i];
    s_bcast = ss;
  }
  __syncthreads();
  e_out[((long long)b * TD_ + d) * TE_ + tid] = ex / s_bcast;
}

// ============================================================================
extern "C" void kernel_launch(void* const* d_in, const int* in_sizes, int n_in,
                              void* d_out, int out_size, void* d_ws, size_t ws_size,
                              hipStream_t stream) {
  const float* enc = (const float*)d_in[0];  // [B,TE,HE]
  const float* dec = (const float*)d_in[1];  // [B,TD,HD]
  const float* Wa  = (const float*)d_in[2];  // [HE,HE]
  const float* Ua  = (const float*)d_in[3];  // [HD,HE]
  const float* Va  = (const float*)d_in[4];  // [HE,1]

  float* c_out = (float*)d_out;                              // [B,TD,HE]
  float* e_out = c_out + (long long)B_ * TD_ * HE_;          // [B,TD,TE]

  float* ws_Ws = (float*)d_ws;                               // [B,TE,HE] 16 MB
  float* ws_Uh = ws_Ws + (long long)B_ * TE_ * HE_;          // [B,TD,HE]  2 MB

  const dim3 wave(32, 1, 1);

  // K1: Ws = enc(8192x512) @ W_a(512x512)   grid 16 x 128
  wmma_gemm512_kernel<<<dim3(LD_ / 32, (B_ * TE_) / 64, 1), wave, 0, stream>>>(
      enc, Wa, ws_Ws, 0, 0, 0);

  // K2: Uh = dec(1024x512) @ U_a(512x512)   grid 16 x 16
  wmma_gemm512_kernel<<<dim3(LD_ / 32, (B_ * TD_) / 64, 1), wave, 0, stream>>>(
      dec, Ua, ws_Uh, 0, 0, 0);

  // K3: fused logits + softmax -> e_out
  logits_softmax_kernel<<<dim3(TD_, B_, 1), dim3(512, 1, 1), 0, stream>>>(
      ws_Ws, ws_Uh, Va, e_out);

  // K4: batched context: c[b] = e[b](64x512) @ enc[b](512x512)   grid 16 x 1 x 16
  wmma_gemm512_kernel<<<dim3(LD_ / 32, TD_ / 64, B_), wave, 0, stream>>>(
      e_out, enc, c_out,
      (long long)TD_ * TE_, (long long)TE_ * HE_, (long long)TD_ * HE_);
}